// EmbeddingNet_7739531067810
// MI455X (gfx1250) — compile-verified
//
#include <hip/hip_runtime.h>
#include <hip/hip_bf16.h>

#define BATCH    64
#define SEQ      5000
#define EMB      128
#define TILES    ((SEQ + 15) / 16)   // 313 row-tiles per batch

// The store-masking trick below requires each lane's 8 rows (s_base+half+0..7)
// to be all-valid or all-invalid, i.e. SEQ % 16 must be 0 or 8.
static_assert(SEQ % 8 == 0, "per-halfwave row-group validity trick requires SEQ % 8 == 0");

typedef __attribute__((ext_vector_type(2))) float v2f;
typedef __attribute__((ext_vector_type(8))) float v8f;

// ---------------------------------------------------------------------------
// Kernel 1: visited_time via log-depth pointer doubling in LDS.
// One workgroup per batch. pack = ptr | (d << 16); node 0 absorbing.
// After 13 rounds (2^13 >= 4999), d[v] = dist(v -> 0); vt[v] = SEQ - d[v].
// ---------------------------------------------------------------------------
__global__ void __launch_bounds__(256)
vt_doubling_kernel(const int* __restrict__ sol, float* __restrict__ vt) {
    __shared__ int pA[SEQ];
    __shared__ int pB[SEQ];
    const int b   = blockIdx.x;
    const int tid = threadIdx.x;
    const int NT  = blockDim.x;
    const int* succ = sol + b * SEQ;

    for (int v = tid; v < SEQ; v += NT) {
        int s = succ[v];                        // 0 <= s < 5000 (fits 16 bits)
        pA[v] = (v == 0) ? 0 : (s | (1 << 16)); // ptr=succ, d=1; node0: ptr=0,d=0
    }
    __syncthreads();

    int* A  = pA;
    int* Bp = pB;
    #pragma unroll 1
    for (int it = 0; it < 13; ++it) {
        for (int v = tid; v < SEQ; v += NT) {
            int pv = A[v];
            int pu = A[pv & 0xffff];
            Bp[v] = (pu & 0xffff) | (((pv >> 16) + (pu >> 16)) << 16);
        }
        __syncthreads();
        int* t = A; A = Bp; Bp = t;
    }

    for (int v = tid; v < SEQ; v += NT) {
        int d = A[v] >> 16;                    // dist(v -> 0), 0..4999
        vt[b * SEQ + v] = (float)(SEQ - d);    // visited_time in 1..SEQ
    }
}

// ---------------------------------------------------------------------------
// Kernel 2: NFE = x @ W^T via V_WMMA_F32_16X16X4_F32 (K=2 zero-padded to 4).
// One wave handles 16 sequence rows x full 128 emb dims (8 N-tiles).
// A layout (f32 16x4): lanes 0-15 = M rows, v0=K0, v1=K1; lanes 16-31 = K2,K3 (0).
// B layout (f32 4x16): lanes 0-15 = N cols, v0=K0, v1=K1; lanes 16-31 zero.
// D layout: VGPR r -> M = r + (lane>=16 ? 8 : 0), N = lane & 15.
// Loads are UNpredicated (clamped addresses) + v_cndmask zeroing. The store
// block is a single fully-unrolled region guarded by ONE per-lane predicate
// (s_base + half < SEQ): since SEQ%16==8, a lane's 8 rows are all-valid or
// all-invalid. Constant indexing only -> accumulators stay in VGPRs (no
// scratch spills), immediate store offsets (r*512B, nb*64B).
// ---------------------------------------------------------------------------
__global__ void __launch_bounds__(256)
nfe_wmma_kernel(const float* __restrict__ x,
                const float* __restrict__ W,
                float* __restrict__ nfe) {
    const int lane = threadIdx.x & 31;
    const int wave = threadIdx.x >> 5;
    const int gw   = blockIdx.x * (blockDim.x >> 5) + wave;   // wave-uniform
    const int b      = gw / TILES;
    const int s_base = (gw % TILES) * 16;

    const int  l15 = lane & 15;
    const bool lo  = (lane < 16);

    // A: 16 x-rows for this tile; all lanes load a clamped in-bounds address,
    // upper half-wave (K=2,3) and rows past SEQ are zeroed by select.
    const int s   = s_base + l15;
    const int sc  = (s < SEQ) ? s : (SEQ - 1);
    const float2 xv = ((const float2*)x)[(size_t)b * SEQ + sc];
    const bool akeep = lo && (s < SEQ);
    v2f a;
    a.x = akeep ? xv.x : 0.0f;
    a.y = akeep ? xv.y : 0.0f;

    // B: all 8 N-tiles of W^T (W is 1KB, L2-hot); unpredicated loads.
    v2f bm[8];
    #pragma unroll
    for (int nb = 0; nb < 8; ++nb) {
        const float2 wv = ((const float2*)W)[nb * 16 + l15];  // W[n][0..1]
        bm[nb].x = lo ? wv.x : 0.0f;
        bm[nb].y = lo ? wv.y : 0.0f;
    }

    const v8f z = {};
    v8f c[8];
    #pragma unroll
    for (int nb = 0; nb < 8; ++nb) {
        c[nb] = __builtin_amdgcn_wmma_f32_16x16x4_f32(
                    false, a, false, bm[nb], (short)0, z, false, false);
    }

    const int half = (lane >> 4) << 3;   // 0 or 8
    float* base = nfe + ((size_t)b * SEQ + (s_base + half)) * EMB + l15;

    // Single predicate covers full tiles (all lanes true) and the last tile
    // of each batch (upper half-wave false). 64 stores, constant offsets.
    if (s_base + half < SEQ) {
        #pragma unroll
        for (int nb = 0; nb < 8; ++nb) {
            float* p = base + nb * 16;
            #pragma unroll
            for (int r = 0; r < 8; ++r)
                p[r * EMB] = c[nb][r];   // immediate offsets: r*512B
        }
    }
}

// ---------------------------------------------------------------------------
// Kernel 3: PFE row gather. 256 threads = 8 rows x 32 float4 lanes.
// pattern (2.56 MB) is L2-resident; stores are the bandwidth cost.
// ---------------------------------------------------------------------------
__global__ void __launch_bounds__(256)
pfe_gather_kernel(const float* __restrict__ pattern,
                  const float* __restrict__ vt,
                  float* __restrict__ pfe) {
    const int row = blockIdx.x * 8 + (threadIdx.x >> 5);   // 0 .. B*SEQ-1
    const int c4  = threadIdx.x & 31;                       // float4 column
    int vti = (int)vt[row];
    int idx = vti % SEQ;                                    // vt==SEQ -> 0
    const float4 v = ((const float4*)pattern)[idx * 32 + c4];
    ((float4*)pfe)[(size_t)row * 32 + c4] = v;
}

// ---------------------------------------------------------------------------
extern "C" void kernel_launch(void* const* d_in, const int* in_sizes, int n_in,
                              void* d_out, int out_size, void* d_ws, size_t ws_size,
                              hipStream_t stream) {
    const float* x       = (const float*)d_in[0];   // [64,5000,2]
    const float* W       = (const float*)d_in[1];   // [128,2]
    const float* pattern = (const float*)d_in[2];   // [5000,128]
    const int*   sol     = (const int*)d_in[3];     // [64,5000]

    float* out = (float*)d_out;
    float* nfe = out;                                        // [64,5000,128]
    float* pfe = out + (size_t)BATCH * SEQ * EMB;            // [64,5000,128]
    float* vt  = out + (size_t)2 * BATCH * SEQ * EMB;        // [64,5000] (as float)

    // 1) visited_time (needed by PFE gather)
    vt_doubling_kernel<<<BATCH, 256, 0, stream>>>(sol, vt);

    // 2) NFE GEMM: 64*313 = 20032 wave-tiles / 8 waves per block = 2504 blocks
    nfe_wmma_kernel<<<(BATCH * TILES) / 8, 256, 0, stream>>>(x, W, nfe);

    // 3) PFE gather: 64*5000 rows / 8 rows per block = 40000 blocks
    pfe_gather_kernel<<<(BATCH * SEQ) / 8, 256, 0, stream>>>(pattern, vt, pfe);
}